// AttentionLayer_1443109011680
// MI455X (gfx1250) — compile-verified
//
#include <hip/hip_runtime.h>

// ---------------------------------------------------------------------------
// Luong attention for MI455X (gfx1250, wave32).
//   S = enc @ dec^T   (per batch 1024^3 fp32 GEMM)            -> workspace
//   softmax over e: per-(b,t) column max & 1/sum              -> workspace
//   ctx = softmax(S)^T @ enc  (exp applied during LDS staging)
//   out = concat(dec, ctx)
// Matrix math via V_WMMA_F32_16X16X4_F32 (exact fp32 path).
// 256x128 block tile, 8 waves x (64x64) = 4x4 accumulators each,
// double-buffered LDS (one barrier per k-chunk), bank-conflict-free pads.
// ---------------------------------------------------------------------------

typedef __attribute__((ext_vector_type(2))) float v2f;
typedef __attribute__((ext_vector_type(8))) float v8f;

#define WMMA_F32(a, b, c)                                                      \
  __builtin_amdgcn_wmma_f32_16x16x4_f32(false, (a), false, (b), (short)0, (c), \
                                        false, false)

constexpr int kB  = 16;
constexpr int kTe = 1024;
constexpr int kTd = 1024;
constexpr int kD  = 1024;

constexpr int BM   = 256;  // block tile M (e or t)
constexpr int BN   = 128;  // block tile N (t or d)
constexpr int BK   = 16;   // k chunk
constexpr int PADA = 4;    // row-major tiles: stride 20 tiles all 64 banks
constexpr int PADC = 8;    // k-major tiles: 2-row offset = 16 mod 64 banks

// ---------------------------------------------------------------------------
// Kernel A: S[b,e,t] = sum_d enc[b,e,d] * dec[b,t,d]
// ---------------------------------------------------------------------------
__global__ __launch_bounds__(256) void score_gemm(const float* __restrict__ enc,
                                                  const float* __restrict__ dec,
                                                  float* __restrict__ S) {
  __shared__ float Alds[2][BM][BK + PADA];  // enc tile [e][k]  (40 KB)
  __shared__ float Blds[2][BN][BK + PADA];  // dec tile [t][k]  (20 KB)

  const int tid   = threadIdx.x;
  const int lane  = tid & 31;
  const int wave  = tid >> 5;
  const int wM    = (wave & 3) * 64;   // e offset inside block tile
  const int wN    = (wave >> 2) * 64;  // t offset inside block tile
  const int b     = blockIdx.z;
  const int tileE = blockIdx.y * BM;
  const int tileT = blockIdx.x * BN;

  const float* encB = enc + (size_t)b * kTe * kD;
  const float* decB = dec + (size_t)b * kTd * kD;

  const int ldr   = tid >> 2;       // 0..63
  const int ldc   = (tid & 3) * 4;  // 0,4,8,12
  const int m16   = lane & 15;
  const int khalf = (lane >> 4) * 2;

  v8f    acc[4][4] = {};
  float4 aR[4], bR[2];

  auto gload = [&](int kk) {
#pragma unroll
    for (int h = 0; h < 4; ++h)
      aR[h] = *(const float4*)(encB + (size_t)(tileE + ldr + h * 64) * kD + kk + ldc);
#pragma unroll
    for (int h = 0; h < 2; ++h)
      bR[h] = *(const float4*)(decB + (size_t)(tileT + ldr + h * 64) * kD + kk + ldc);
  };
  auto stage = [&](int buf) {
#pragma unroll
    for (int h = 0; h < 4; ++h) *(float4*)&Alds[buf][ldr + h * 64][ldc] = aR[h];
#pragma unroll
    for (int h = 0; h < 2; ++h) *(float4*)&Blds[buf][ldr + h * 64][ldc] = bR[h];
  };
  auto mma_chunk = [&](int buf) {
#pragma unroll
    for (int ks = 0; ks < BK; ks += 4) {
      v2f af[4], bf[4];
#pragma unroll
      for (int mi = 0; mi < 4; ++mi) {
        const float* p = &Alds[buf][wM + mi * 16 + m16][ks + khalf];
        af[mi].x = p[0];
        af[mi].y = p[1];
      }
#pragma unroll
      for (int ni = 0; ni < 4; ++ni) {
        const float* p = &Blds[buf][wN + ni * 16 + m16][ks + khalf];
        bf[ni].x = p[0];
        bf[ni].y = p[1];
      }
#pragma unroll
      for (int mi = 0; mi < 4; ++mi)
#pragma unroll
        for (int ni = 0; ni < 4; ++ni)
          acc[mi][ni] = WMMA_F32(af[mi], bf[ni], acc[mi][ni]);
    }
  };

  gload(0);
  stage(0);
  __syncthreads();
  int cur = 0;
  for (int kk = BK; kk < kD; kk += BK) {
    gload(kk);        // next chunk in flight while we compute
    mma_chunk(cur);   // 64 WMMAs on current buffer
    stage(cur ^ 1);   // waits loadcnt, fills other buffer
    __syncthreads();
    cur ^= 1;
  }
  mma_chunk(cur);

  const int rowHi = (lane >> 4) << 3;
#pragma unroll
  for (int mi = 0; mi < 4; ++mi)
#pragma unroll
    for (int ni = 0; ni < 4; ++ni) {
      const int t = tileT + wN + ni * 16 + m16;
#pragma unroll
      for (int r = 0; r < 8; ++r) {
        const int e = tileE + wM + mi * 16 + rowHi + r;
        S[((size_t)b * kTe + e) * kTd + t] = acc[mi][ni][r];
      }
    }
}

// ---------------------------------------------------------------------------
// Kernel B: per-(b,t) online softmax stats over e (coalesced across t).
// ---------------------------------------------------------------------------
__global__ __launch_bounds__(256) void softmax_stats(const float* __restrict__ S,
                                                     float* __restrict__ stats) {
  const int c = blockIdx.x * 256 + threadIdx.x;  // 0 .. B*Td-1
  const int b = c / kTd;
  const int t = c - b * kTd;
  const float* col = S + (size_t)b * kTe * kTd + t;

  float m = -INFINITY, s = 0.0f;
  for (int e = 0; e < kTe; ++e) {
    const float x  = col[(size_t)e * kTd];
    const float mn = fmaxf(m, x);
    s = s * __expf(m - mn) + __expf(x - mn);
    m = mn;
  }
  stats[2 * c]     = m;
  stats[2 * c + 1] = 1.0f / s;
}

// ---------------------------------------------------------------------------
// Kernel C: ctx[b,t,d] = sum_e P[e,t]*enc[e,d],  P = exp(S - max_t)*inv_sum_t
// (exp applied during staging). Both operands are e(k)-major in memory.
// ---------------------------------------------------------------------------
__global__ __launch_bounds__(256) void context_gemm(const float* __restrict__ S,
                                                    const float* __restrict__ stats,
                                                    const float* __restrict__ enc,
                                                    float* __restrict__ out) {
  __shared__ float Plds[2][BK][BM + PADC];  // [e][t]  (33.8 KB)
  __shared__ float Elds[2][BK][BN + PADC];  // [e][d]  (17.4 KB)
  __shared__ float sMax[BM];
  __shared__ float sInv[BM];

  const int tid   = threadIdx.x;
  const int lane  = tid & 31;
  const int wave  = tid >> 5;
  const int wM    = (wave & 3) * 64;   // t offset
  const int wN    = (wave >> 2) * 64;  // d offset
  const int b     = blockIdx.z;
  const int tileT = blockIdx.y * BM;
  const int tileD = blockIdx.x * BN;

  const float* Sb   = S + (size_t)b * kTe * kTd;
  const float* encB = enc + (size_t)b * kTe * kD;

  {  // per-column softmax stats for this block's 256 t values
    const int c = b * kTd + tileT + tid;
    sMax[tid]   = stats[2 * c];
    sInv[tid]   = stats[2 * c + 1];
  }
  __syncthreads();

  const int pr    = tid >> 4;         // 0..15  (e row within chunk)
  const int pcP   = (tid & 15) * 16;  // P cols: 16 per thread
  const int pcE   = (tid & 15) * 8;   // E cols: 8 per thread
  const int m16   = lane & 15;
  const int khalf = (lane >> 4) * 2;

  v8f    acc[4][4] = {};
  float4 pR[4], eR[2];

  auto gload = [&](int kk) {
#pragma unroll
    for (int h = 0; h < 4; ++h)
      pR[h] = *(const float4*)(Sb + (size_t)(kk + pr) * kTd + tileT + pcP + h * 4);
#pragma unroll
    for (int h = 0; h < 2; ++h)
      eR[h] = *(const float4*)(encB + (size_t)(kk + pr) * kD + tileD + pcE + h * 4);
  };
  auto stage = [&](int buf) {
#pragma unroll
    for (int h = 0; h < 4; ++h) {
      const int c0 = pcP + h * 4;
      float4    v  = pR[h];
      v.x = __expf(v.x - sMax[c0 + 0]) * sInv[c0 + 0];
      v.y = __expf(v.y - sMax[c0 + 1]) * sInv[c0 + 1];
      v.z = __expf(v.z - sMax[c0 + 2]) * sInv[c0 + 2];
      v.w = __expf(v.w - sMax[c0 + 3]) * sInv[c0 + 3];
      *(float4*)&Plds[buf][pr][c0] = v;
    }
#pragma unroll
    for (int h = 0; h < 2; ++h) *(float4*)&Elds[buf][pr][pcE + h * 4] = eR[h];
  };
  auto mma_chunk = [&](int buf) {
#pragma unroll
    for (int ks = 0; ks < BK; ks += 4) {
      v2f af[4], bf[4];
#pragma unroll
      for (int mi = 0; mi < 4; ++mi) {
        af[mi].x = Plds[buf][ks + khalf][wM + mi * 16 + m16];  // A'[t][e]=P[e][t]
        af[mi].y = Plds[buf][ks + khalf + 1][wM + mi * 16 + m16];
      }
#pragma unroll
      for (int ni = 0; ni < 4; ++ni) {
        bf[ni].x = Elds[buf][ks + khalf][wN + ni * 16 + m16];  // B'[e][d]
        bf[ni].y = Elds[buf][ks + khalf + 1][wN + ni * 16 + m16];
      }
#pragma unroll
      for (int mi = 0; mi < 4; ++mi)
#pragma unroll
        for (int ni = 0; ni < 4; ++ni)
          acc[mi][ni] = WMMA_F32(af[mi], bf[ni], acc[mi][ni]);
    }
  };

  gload(0);
  stage(0);
  __syncthreads();
  int cur = 0;
  for (int kk = BK; kk < kTe; kk += BK) {
    gload(kk);
    mma_chunk(cur);
    stage(cur ^ 1);
    __syncthreads();
    cur ^= 1;
  }
  mma_chunk(cur);

  const int rowHi = (lane >> 4) << 3;
#pragma unroll
  for (int mi = 0; mi < 4; ++mi)
#pragma unroll
    for (int ni = 0; ni < 4; ++ni) {
      const int d = tileD + wN + ni * 16 + m16;
#pragma unroll
      for (int r = 0; r < 8; ++r) {
        const int t = tileT + wM + mi * 16 + rowHi + r;
        out[((size_t)b * kTd + t) * (2 * kD) + kD + d] = acc[mi][ni][r];
      }
    }
}

// ---------------------------------------------------------------------------
// Kernel D: out[b,t,0:D] = dec[b,t,:]  (float4 coalesced copy)
// ---------------------------------------------------------------------------
__global__ __launch_bounds__(256) void copy_dec(const float4* __restrict__ dec4,
                                                float* __restrict__ out) {
  const size_t i4   = (size_t)blockIdx.x * 256 + threadIdx.x;
  const size_t elem = i4 * 4;
  const size_t row  = elem / kD;  // b*Td + t
  const int    d    = (int)(elem - row * kD);
  *(float4*)&out[row * (size_t)(2 * kD) + d] = dec4[i4];
}

// ---------------------------------------------------------------------------
extern "C" void kernel_launch(void* const* d_in, const int* in_sizes, int n_in,
                              void* d_out, int out_size, void* d_ws, size_t ws_size,
                              hipStream_t stream) {
  (void)in_sizes; (void)n_in; (void)out_size; (void)ws_size;
  const float* enc = (const float*)d_in[0];
  const float* dec = (const float*)d_in[1];
  float*       out = (float*)d_out;

  float* S     = (float*)d_ws;                // B*Te*Td fp32 (~67 MB)
  float* stats = S + (size_t)kB * kTe * kTd;  // B*Td * {max, 1/sum}

  score_gemm<<<dim3(kTd / BN, kTe / BM, kB), 256, 0, stream>>>(enc, dec, S);
  softmax_stats<<<(kB * kTd) / 256, 256, 0, stream>>>(S, stats);
  context_gemm<<<dim3(kD / BN, kTd / BM, kB), 256, 0, stream>>>(S, stats, enc, out);
  copy_dec<<<(kB * kTd * kD / 4) / 256, 256, 0, stream>>>((const float4*)dec, out);
}